// DilationSumLayer_35897336660640
// MI455X (gfx1250) — compile-verified
//
#include <hip/hip_runtime.h>
#include <stdint.h>

// -------- problem constants --------
#define IMG_H 512
#define IMG_W 512
#define KH 7
#define KW 7
#define PAD 3

// -------- tiling --------
#define TILE_H 32
#define TILE_W 64
#define HALO_H (TILE_H + KH - 1)   // 38
#define HALO_W (TILE_W + KW - 1)   // 70
#define LDS_PITCH HALO_W           // TDM writes tile rows contiguously
#define BLOCK 256                  // 8 waves (wave32)
#define ROWS_PER_WAVE 5            // 8 waves x 5 >= 38

// -------- CDNA5 feature guards --------
#ifndef __has_builtin
#define __has_builtin(x) 0
#endif

#if __has_builtin(__builtin_amdgcn_tensor_load_to_lds) && \
    __has_builtin(__builtin_amdgcn_s_wait_tensorcnt)
#define HAVE_TDM 1
#else
#define HAVE_TDM 0
#endif

#if __has_builtin(__builtin_amdgcn_global_load_async_to_lds_b32)
#define HAVE_ASYNC_LDS 1
#else
#define HAVE_ASYNC_LDS 0
#endif

typedef __attribute__((address_space(1))) int* as1_int;   // global
typedef __attribute__((address_space(3))) int* as3_int;   // LDS

typedef uint32_t u32x4 __attribute__((ext_vector_type(4)));
typedef int      i32x8 __attribute__((ext_vector_type(8)));
typedef int      i32x4 __attribute__((ext_vector_type(4)));

__device__ __forceinline__ void wait_async0() {
#if __has_builtin(__builtin_amdgcn_s_wait_asynccnt)
    __builtin_amdgcn_s_wait_asynccnt(0);
#else
    asm volatile("s_wait_asynccnt 0" ::: "memory");
#endif
}

__global__ __launch_bounds__(BLOCK) void dilate7x7_kernel(
    const float* __restrict__ x,      // (64, 512, 512) flattened B*C
    const float* __restrict__ w,      // (7,7)
    float* __restrict__ out)          // (64, 512, 512)
{
    __shared__ float tile[HALO_H * LDS_PITCH];

    const int tid = threadIdx.x;
    const int tx  = blockIdx.x;       // 0..7   (512/64)
    const int ty  = blockIdx.y;       // 0..15  (512/32)
    const int img = blockIdx.z;       // 0..63  (B*C)

    const float* src = x + (size_t)img * (IMG_H * IMG_W);

    const int gx0 = tx * TILE_W - PAD;   // leftmost input col of halo tile
    const int gy0 = ty * TILE_H - PAD;   // topmost input row of halo tile

    const bool interior = (gx0 >= 0) && (gy0 >= 0) &&
                          (gx0 + HALO_W <= IMG_W) && (gy0 + HALO_H <= IMG_H);

    if (interior) {
#if HAVE_TDM
        // Tensor Data Mover: each wave DMAs a <=5-row slice of the 70x38 f32
        // halo tile straight into LDS. One instruction per wave; no per-lane
        // address VALU; tracked with TENSORcnt.
        {
            const int wave  = __builtin_amdgcn_readfirstlane((int)(tid >> 5));
            const int row0  = wave * ROWS_PER_WAVE;                  // 0..35
            const int nrows = (HALO_H - row0) < ROWS_PER_WAVE ?
                              (HALO_H - row0) : ROWS_PER_WAVE;       // 5 or 3
            const uint64_t gaddr =
                (uint64_t)(uintptr_t)(src + (size_t)(gy0 + row0) * IMG_W + gx0);
            const uint32_t laddr =
                (uint32_t)(uintptr_t)&tile[row0 * LDS_PITCH];

            // ---- D# group 0 (128b): control | lds_addr | global_addr | type=2
            u32x4 g0;
            g0.x = 1u;                                   // count=1, user mode
            g0.y = laddr;                                // lds_addr (bytes)
            g0.z = (uint32_t)gaddr;                      // global_addr[31:0]
            g0.w = (uint32_t)(gaddr >> 32) | (2u << 30); // addr[56:32] | type=2

            // ---- D# group 1 (256b): sizes/strides (2D tile, 4B elements)
            i32x8 g1;
            g1[0] = (2 << 16);                           // data_size=4B; mask=0
            g1[1] = (int)(((uint32_t)IMG_W & 0xFFFFu) << 16);        // dim0 lo
            g1[2] = (int)(((uint32_t)IMG_W >> 16) |
                          (((uint32_t)IMG_H & 0xFFFFu) << 16));      // dim0 hi | dim1 lo
            g1[3] = (int)(((uint32_t)IMG_H >> 16) |
                          ((uint32_t)HALO_W << 16));     // dim1 hi | tile_dim0=70
            g1[4] = nrows & 0xFFFF;                      // tile_dim1 | tile_dim2=0
            g1[5] = IMG_W;                               // dim0_stride lo32 = 512
            g1[6] = 0;                                   // dim0_stride hi | dim1_stride lo
            g1[7] = 0;                                   // dim1_stride hi (unused, 2D)

            // ---- D# groups 2/3 + trailing group: zero (tensor is 2D)
            i32x4 g2 = {0, 0, 0, 0};
            i32x4 g3 = {0, 0, 0, 0};
            i32x8 g4 = {0, 0, 0, 0, 0, 0, 0, 0};

            __builtin_amdgcn_tensor_load_to_lds(g0, g1, g2, g3, g4, /*cpol*/0);
            __builtin_amdgcn_s_wait_tensorcnt(0);
        }
#elif HAVE_ASYNC_LDS
        for (int idx = tid; idx < HALO_H * HALO_W; idx += BLOCK) {
            const int r = idx / HALO_W;
            const int c = idx - r * HALO_W;
            const float* g = src + (size_t)(gy0 + r) * IMG_W + (gx0 + c);
            float*       l = &tile[r * LDS_PITCH + c];
            __builtin_amdgcn_global_load_async_to_lds_b32(
                (as1_int)(uintptr_t)g,
                (as3_int)(uint32_t)(uintptr_t)l,
                /*imm offset*/0, /*cpol*/0);
        }
        wait_async0();
#else
        for (int idx = tid; idx < HALO_H * HALO_W; idx += BLOCK) {
            const int r = idx / HALO_W;
            const int c = idx - r * HALO_W;
            tile[r * LDS_PITCH + c] = src[(size_t)(gy0 + r) * IMG_W + (gx0 + c)];
        }
#endif
    } else {
        // Boundary tile: reference zero-pads, so OOB taps read 0.0f.
        for (int idx = tid; idx < HALO_H * HALO_W; idx += BLOCK) {
            const int r  = idx / HALO_W;
            const int c  = idx - r * HALO_W;
            const int gy = gy0 + r;
            const int gx = gx0 + c;
            float v = 0.0f;
            if (gy >= 0 && gy < IMG_H && gx >= 0 && gx < IMG_W)
                v = src[(size_t)gy * IMG_W + gx];
            tile[r * LDS_PITCH + c] = v;
        }
    }
    __syncthreads();

    // ---- compute: 8 output rows per thread, one column ----
    const int j  = tid & (TILE_W - 1);        // 0..63
    const int r0 = (tid >> 6) * 8;            // 0,8,16,24

    float acc[8];
#pragma unroll
    for (int r = 0; r < 8; ++r) acc[r] = -__builtin_inff();

#pragma unroll
    for (int kx = 0; kx < KW; ++kx) {
        // One 14-deep LDS column feeds all 7 kernel rows x 8 output rows.
        float v[8 + KH - 1];
#pragma unroll
        for (int y = 0; y < 8 + KH - 1; ++y)
            v[y] = tile[(r0 + y) * LDS_PITCH + j + kx];
#pragma unroll
        for (int ky = 0; ky < KH; ++ky) {
            const float wv = w[ky * KW + kx];   // uniform -> s_load, hoisted
#pragma unroll
            for (int r = 0; r < 8; ++r)
                acc[r] = fmaxf(acc[r], v[r + ky] + wv);
        }
    }

    float* dst = out + (size_t)img * (IMG_H * IMG_W)
                     + (size_t)(ty * TILE_H + r0) * IMG_W
                     + (tx * TILE_W + j);
#pragma unroll
    for (int r = 0; r < 8; ++r)
        dst[(size_t)r * IMG_W] = acc[r];      // coalesced along j
}

extern "C" void kernel_launch(void* const* d_in, const int* in_sizes, int n_in,
                              void* d_out, int out_size, void* d_ws, size_t ws_size,
                              hipStream_t stream) {
    (void)in_sizes; (void)n_in; (void)d_ws; (void)ws_size; (void)out_size;
    const float* x = (const float*)d_in[0];   // (8,8,512,512) f32
    const float* w = (const float*)d_in[1];   // (7,7) f32
    float* out = (float*)d_out;

    dim3 grid(IMG_W / TILE_W, IMG_H / TILE_H, 64);  // (8, 16, B*C)
    dim3 block(BLOCK, 1, 1);
    dilate7x7_kernel<<<grid, block, 0, stream>>>(x, w, out);
}